// SpatialTemporalGNN_12111807775254
// MI455X (gfx1250) — compile-verified
//
#include <hip/hip_runtime.h>
#include <hip/hip_bf16.h>

typedef __attribute__((ext_vector_type(2))) float v2f;
typedef __attribute__((ext_vector_type(8))) float v8f;

#define N_NODES   50000
#define N_EDGES   800000
#define IN_C      128
#define HID_C     256
#define OUT_C     128
#define NUM_GRAPHS 128
#define BN_EPS    1e-5f

// ---------------------------------------------------------------------------
// Generic zero-fill (graph-capture safe; we avoid hipMemsetAsync ambiguity)
// ---------------------------------------------------------------------------
__global__ void zero_f32(float* __restrict__ p, unsigned long long n) {
    unsigned long long i = (unsigned long long)blockIdx.x * blockDim.x + threadIdx.x;
    if (i < n) p[i] = 0.0f;
}

// ---------------------------------------------------------------------------
// Degree (with self loops) -> dinv = rsqrt(deg+1)
// ---------------------------------------------------------------------------
__global__ void compute_deg(const int* __restrict__ dst, float* __restrict__ deg, int E) {
    int e = blockIdx.x * blockDim.x + threadIdx.x;
    if (e < E) atomicAdd(&deg[dst[e]], 1.0f);
}

__global__ void compute_dinv(float* __restrict__ deg, int N) {
    int i = blockIdx.x * blockDim.x + threadIdx.x;
    if (i < N) deg[i] = rsqrtf(deg[i] + 1.0f);   // +1 for the self loop; always > 0
}

// ---------------------------------------------------------------------------
// f32 GEMM via V_WMMA_F32_16X16X4_F32.
// One wave computes a 16(row) x 64(col) output tile: 4 accumulators, A frag
// shared across the 4 WMMAs per K-step.
// A-matrix 16x4 f32 layout (ISA 7.12.2): lanes 0-15 hold M=lane, K={k,k+1};
// lanes 16-31 hold M=lane-16, K={k+2,k+3}. B mirrors with N across lanes.
// C/D: VGPR r -> row (r + 8*hi), col = lane&15.
// ---------------------------------------------------------------------------
__global__ void wmma_gemm_f32(const float* __restrict__ A, const float* __restrict__ W,
                              float* __restrict__ Out, int N, int K, int C) {
    const int lane  = threadIdx.x & 31;
    const int gwave = blockIdx.x * (blockDim.x >> 5) + (threadIdx.x >> 5);
    const int colGroups = C >> 6;                 // 64-wide column groups
    const int rowTiles  = N >> 4;
    if (gwave >= rowTiles * colGroups) return;
    const int rowTile = gwave / colGroups;
    const int colGrp  = gwave % colGroups;

    const int m  = lane & 15;
    const int hi = lane >> 4;                     // 0: K lo pair, 1: K hi pair
    const int row = rowTile * 16 + m;
    const int colBase = colGrp * 64 + m;

    v8f acc0 = {}, acc1 = {}, acc2 = {}, acc3 = {};
    const float* __restrict__ arow = A + (size_t)row * K;

    for (int k = 0; k < K; k += 4) {
        const int ka = k + (hi << 1);
        v2f a;
        a.x = arow[ka];
        a.y = arow[ka + 1];
        const float* __restrict__ w0 = W + (size_t)ka * C;
        const float* __restrict__ w1 = W + (size_t)(ka + 1) * C;
        v2f b0, b1, b2, b3;
        b0.x = w0[colBase +  0]; b0.y = w1[colBase +  0];
        b1.x = w0[colBase + 16]; b1.y = w1[colBase + 16];
        b2.x = w0[colBase + 32]; b2.y = w1[colBase + 32];
        b3.x = w0[colBase + 48]; b3.y = w1[colBase + 48];
        acc0 = __builtin_amdgcn_wmma_f32_16x16x4_f32(false, a, false, b0, (short)0, acc0, false, false);
        acc1 = __builtin_amdgcn_wmma_f32_16x16x4_f32(false, a, false, b1, (short)0, acc1, false, false);
        acc2 = __builtin_amdgcn_wmma_f32_16x16x4_f32(false, a, false, b2, (short)0, acc2, false, false);
        acc3 = __builtin_amdgcn_wmma_f32_16x16x4_f32(false, a, false, b3, (short)0, acc3, false, false);
    }

    const int orow = rowTile * 16 + hi * 8;
    float* __restrict__ ob = Out + (size_t)orow * C + colGrp * 64 + m;
    #pragma unroll
    for (int r = 0; r < 8; ++r) {
        ob[(size_t)r * C +  0] = acc0[r];
        ob[(size_t)r * C + 16] = acc1[r];
        ob[(size_t)r * C + 32] = acc2[r];
        ob[(size_t)r * C + 48] = acc3[r];
    }
}

// ---------------------------------------------------------------------------
// GCN aggregation: agg[dst] += lin[src] * dinv[src]*dinv[dst] over edges,
// plus self loops (edge index >= E -> src=dst=i, norm=dinv[i]^2).
// One wave per edge; lanes sweep channels (coalesced 128B bursts, L2-resident).
// ---------------------------------------------------------------------------
__global__ void gcn_aggregate(const float* __restrict__ lin, float* __restrict__ agg,
                              const int* __restrict__ src, const int* __restrict__ dst,
                              const float* __restrict__ dinv, int E, int N, int C) {
    const int wave = (blockIdx.x * blockDim.x + threadIdx.x) >> 5;
    const int lane = threadIdx.x & 31;
    if (wave >= E + N) return;
    int s, d;
    if (wave < E) { s = src[wave]; d = dst[wave]; }
    else          { s = d = wave - E; }
    const float nrm = dinv[s] * dinv[d];
    const float* __restrict__ lp = lin + (size_t)s * C;
    float* __restrict__ op = agg + (size_t)d * C;
    for (int c = lane; c < C; c += 32)
        atomicAdd(&op[c], lp[c] * nrm);
}

// ---------------------------------------------------------------------------
// BN stats: blockDim.x == C; each thread owns one channel, blocks stride rows
// (fully coalesced row-major reads). Partial sums reduced with 2 atomics/chan.
// stats[0..C) = sum, stats[C..2C) = sum of squares.
// ---------------------------------------------------------------------------
__global__ void bn_stats(const float* __restrict__ h, float* __restrict__ stats,
                         int N, int C) {
    const int c = threadIdx.x;
    float s = 0.0f, s2 = 0.0f;
    for (int r = blockIdx.x; r < N; r += gridDim.x) {
        const float v = h[(size_t)r * C + c];
        s  += v;
        s2 += v * v;
    }
    atomicAdd(&stats[c],     s);
    atomicAdd(&stats[C + c], s2);
}

__global__ void bn_relu(float* __restrict__ h, const float* __restrict__ stats,
                        const float* __restrict__ gamma, const float* __restrict__ beta,
                        int N, int C) {
    const unsigned long long idx =
        (unsigned long long)blockIdx.x * blockDim.x + threadIdx.x;
    if (idx >= (unsigned long long)N * C) return;
    const int c = (int)(idx & (unsigned)(C - 1));     // C is a power of two
    const float invN = 1.0f / (float)N;
    const float mu  = stats[c] * invN;
    const float var = stats[C + c] * invN - mu * mu;  // biased var
    float v = (h[idx] - mu) * rsqrtf(var + BN_EPS) * gamma[c] + beta[c];
    h[idx] = v > 0.0f ? v : 0.0f;
}

// ---------------------------------------------------------------------------
// Global mean pool
// ---------------------------------------------------------------------------
__global__ void pool_accum(const float* __restrict__ h, const int* __restrict__ batch,
                           float* __restrict__ sums, float* __restrict__ counts,
                           int N, int C) {
    const int node = (blockIdx.x * blockDim.x + threadIdx.x) >> 5;
    const int lane = threadIdx.x & 31;
    if (node >= N) return;
    const int g = batch[node];
    const float* __restrict__ hp = h + (size_t)node * C;
    float* __restrict__ sp = sums + (size_t)g * C;
    for (int c = lane; c < C; c += 32)
        atomicAdd(&sp[c], hp[c]);
    if (lane == 0) atomicAdd(&counts[g], 1.0f);
}

__global__ void pool_final(const float* __restrict__ sums, const float* __restrict__ counts,
                           float* __restrict__ out, int G, int C) {
    const int idx = blockIdx.x * blockDim.x + threadIdx.x;
    if (idx >= G * C) return;
    const int g = idx / C;
    out[idx] = sums[idx] / fmaxf(counts[g], 1.0f);
}

// ---------------------------------------------------------------------------
// Orchestration
// ---------------------------------------------------------------------------
static inline unsigned int cdiv(unsigned long long a, unsigned int b) {
    return (unsigned int)((a + b - 1) / b);
}

extern "C" void kernel_launch(void* const* d_in, const int* in_sizes, int n_in,
                              void* d_out, int out_size, void* d_ws, size_t ws_size,
                              hipStream_t stream) {
    (void)in_sizes; (void)n_in; (void)out_size; (void)ws_size;

    const float* x     = (const float*)d_in[0];
    const int*   ei    = (const int*)d_in[1];         // [2, E]: src row then dst row
    const int*   batch = (const int*)d_in[2];
    const float* W1 = (const float*)d_in[3];
    const float* W2 = (const float*)d_in[5];
    const float* W3 = (const float*)d_in[7];
    const float* g1 = (const float*)d_in[9],  *be1 = (const float*)d_in[10];
    const float* g2 = (const float*)d_in[11], *be2 = (const float*)d_in[12];
    const float* g3 = (const float*)d_in[13], *be3 = (const float*)d_in[14];
    // b1/b2/b3 are mathematically cancelled by the following BatchNorm -> skipped.

    const int* src = ei;
    const int* dst = ei + N_EDGES;

    float* ws = (float*)d_ws;
    const size_t featMax = (size_t)N_NODES * HID_C;   // 12.8M floats
    float* bufA   = ws;                               // linear output
    float* bufB   = bufA + featMax;                   // aggregated / activated
    float* dinv   = bufB + featMax;                   // N floats (deg -> dinv)
    float* stats  = dinv + N_NODES;                   // 2*HID_C floats
    float* psums  = stats + 2 * HID_C;                // G*OUT_C
    float* pcnts  = psums + (size_t)NUM_GRAPHS * OUT_C;

    const int TB = 256;
    const int wavesPerBlk = TB / 32;
    const int aggBlocks = cdiv((unsigned long long)(N_EDGES + N_NODES), wavesPerBlk);

    // --- dinv ---
    zero_f32<<<cdiv(N_NODES, TB), TB, 0, stream>>>(dinv, N_NODES);
    compute_deg<<<cdiv(N_EDGES, TB), TB, 0, stream>>>(dst, dinv, N_EDGES);
    compute_dinv<<<cdiv(N_NODES, TB), TB, 0, stream>>>(dinv, N_NODES);

    // --- Layer 1: x[N,128] @ W1[128,256] -> aggregate -> BN+ReLU ---
    {
        const int K = IN_C, C = HID_C;
        const int tiles = (N_NODES / 16) * (C / 64);
        wmma_gemm_f32<<<cdiv(tiles, wavesPerBlk), TB, 0, stream>>>(x, W1, bufA, N_NODES, K, C);
        zero_f32<<<cdiv((unsigned long long)N_NODES * C, TB), TB, 0, stream>>>(bufB, (unsigned long long)N_NODES * C);
        gcn_aggregate<<<aggBlocks, TB, 0, stream>>>(bufA, bufB, src, dst, dinv, N_EDGES, N_NODES, C);
        zero_f32<<<cdiv(2 * C, TB), TB, 0, stream>>>(stats, 2 * C);
        bn_stats<<<512, C, 0, stream>>>(bufB, stats, N_NODES, C);
        bn_relu<<<cdiv((unsigned long long)N_NODES * C, TB), TB, 0, stream>>>(bufB, stats, g1, be1, N_NODES, C);
    }

    // --- Layer 2: h1[N,256] @ W2[256,256] ---
    {
        const int K = HID_C, C = HID_C;
        const int tiles = (N_NODES / 16) * (C / 64);
        wmma_gemm_f32<<<cdiv(tiles, wavesPerBlk), TB, 0, stream>>>(bufB, W2, bufA, N_NODES, K, C);
        zero_f32<<<cdiv((unsigned long long)N_NODES * C, TB), TB, 0, stream>>>(bufB, (unsigned long long)N_NODES * C);
        gcn_aggregate<<<aggBlocks, TB, 0, stream>>>(bufA, bufB, src, dst, dinv, N_EDGES, N_NODES, C);
        zero_f32<<<cdiv(2 * C, TB), TB, 0, stream>>>(stats, 2 * C);
        bn_stats<<<512, C, 0, stream>>>(bufB, stats, N_NODES, C);
        bn_relu<<<cdiv((unsigned long long)N_NODES * C, TB), TB, 0, stream>>>(bufB, stats, g2, be2, N_NODES, C);
    }

    // --- Layer 3: h2[N,256] @ W3[256,128] ---
    {
        const int K = HID_C, C = OUT_C;
        const int tiles = (N_NODES / 16) * (C / 64);
        wmma_gemm_f32<<<cdiv(tiles, wavesPerBlk), TB, 0, stream>>>(bufB, W3, bufA, N_NODES, K, C);
        zero_f32<<<cdiv((unsigned long long)N_NODES * C, TB), TB, 0, stream>>>(bufB, (unsigned long long)N_NODES * C);
        gcn_aggregate<<<aggBlocks, TB, 0, stream>>>(bufA, bufB, src, dst, dinv, N_EDGES, N_NODES, C);
        zero_f32<<<cdiv(2 * C, TB), TB, 0, stream>>>(stats, 2 * C);
        bn_stats<<<512, C, 0, stream>>>(bufB, stats, N_NODES, C);
        bn_relu<<<cdiv((unsigned long long)N_NODES * C, TB), TB, 0, stream>>>(bufB, stats, g3, be3, N_NODES, C);
    }

    // --- Global mean pool ---
    {
        const unsigned long long pn = (unsigned long long)NUM_GRAPHS * OUT_C + NUM_GRAPHS;
        zero_f32<<<cdiv(pn, TB), TB, 0, stream>>>(psums, pn);   // sums then counts (contiguous)
        pool_accum<<<cdiv(N_NODES, wavesPerBlk), TB, 0, stream>>>(bufB, batch, psums, pcnts, N_NODES, OUT_C);
        pool_final<<<cdiv(NUM_GRAPHS * OUT_C, TB), TB, 0, stream>>>(psums, pcnts, (float*)d_out, NUM_GRAPHS, OUT_C);
    }
}